// DirGNN_GatConv_model_70042326663393
// MI455X (gfx1250) — compile-verified
//
#include <hip/hip_runtime.h>
#include <hip/hip_bf16.h>
#include <math.h>

#define F 128
#define G_NUM 250

typedef __attribute__((ext_vector_type(2))) float v2f;
typedef __attribute__((ext_vector_type(8))) float v8f;

static __device__ __forceinline__ void atomicMaxFloat(float* addr, float v) {
  if (v >= 0.0f) {
    atomicMax((int*)addr, __float_as_int(v));
  } else {
    atomicMin((unsigned int*)addr, __float_as_uint(v));
  }
}

// ---------------- fill ----------------
__global__ void fill_kernel(float* p, float v, int n) {
  int i = blockIdx.x * blockDim.x + threadIdx.x;
  if (i < n) p[i] = v;
}

// ---------------- degrees ----------------
__global__ void degree_kernel(const int* __restrict__ src, const int* __restrict__ dst,
                              float* din, float* dou, int E) {
  int e = blockIdx.x * blockDim.x + threadIdx.x;
  if (e >= E) return;
  atomicAdd(din + dst[e], 1.0f);
  atomicAdd(dou + src[e], 1.0f);
}

__global__ void dinv_kernel(float* din, float* dou, int N) {
  int i = blockIdx.x * blockDim.x + threadIdx.x;
  if (i >= N) return;
  float a = din[i];  din[i] = (a > 0.f) ? (1.0f / sqrtf(a)) : 0.f;
  float b = dou[i];  dou[i] = (b > 0.f) ? (1.0f / sqrtf(b)) : 0.f;
}

// ---------------- fp32 WMMA GEMM: C[M,128] = A[M,128] @ W[128,128] ----------------
// Block = 256 threads = 8 waves. Block tile: 64 rows x 128 cols.
// Wave w owns cols [16w,16w+16); iterates 4 row tiles of 16.
__global__ __launch_bounds__(256) void gemm128_wmma(
    const float* __restrict__ A, const float* __restrict__ W,
    float* __restrict__ C, int M) {
  __shared__ float sA[64 * 32];
  __shared__ float sW[32 * F];
  const int tid  = threadIdx.x;
  const int wave = tid >> 5;
  const int lane = tid & 31;
  const int half = lane >> 4;      // lanes 16-31 -> K offset +2 (A/B), M offset +8 (C/D)
  const int mrow = lane & 15;
  const int koff = half * 2;
  const int row0 = blockIdx.x * 64;

  v8f acc[4];
  for (int r = 0; r < 4; ++r)
    for (int i = 0; i < 8; ++i) acc[r][i] = 0.0f;

  for (int kb = 0; kb < 4; ++kb) {
    // stage A tile 64x32 (512 float4, 2 per thread)
    for (int i = 0; i < 2; ++i) {
      int idx = tid + i * 256;
      int r = idx >> 3;
      int c = (idx & 7) * 4;
      int grow = row0 + r;
      float4 v = make_float4(0.f, 0.f, 0.f, 0.f);
      if (grow < M) v = *(const float4*)(A + (size_t)grow * F + kb * 32 + c);
      *(float4*)(sA + r * 32 + c) = v;
    }
    // stage W tile 32x128 (1024 float4, 4 per thread)
    for (int i = 0; i < 4; ++i) {
      int idx = tid + i * 256;
      int r = idx >> 5;
      int c = (idx & 31) * 4;
      *(float4*)(sW + r * F + c) = *(const float4*)(W + (size_t)(kb * 32 + r) * F + c);
    }
    __syncthreads();
    for (int ks = 0; ks < 8; ++ks) {
      int k0 = ks * 4 + koff;
      v2f b;
      b[0] = sW[k0 * F + wave * 16 + mrow];
      b[1] = sW[(k0 + 1) * F + wave * 16 + mrow];
      for (int rt = 0; rt < 4; ++rt) {
        const float* ap = sA + (rt * 16 + mrow) * 32 + k0;
        v2f a; a[0] = ap[0]; a[1] = ap[1];
        acc[rt] = __builtin_amdgcn_wmma_f32_16x16x4_f32(
            false, a, false, b, (short)0, acc[rt], false, false);
      }
    }
    __syncthreads();
  }
  for (int rt = 0; rt < 4; ++rt) {
    for (int i = 0; i < 8; ++i) {
      int grow = row0 + rt * 16 + i + half * 8;
      if (grow < M) C[(size_t)grow * F + wave * 16 + mrow] = acc[rt][i];
    }
  }
}

// ---------------- GCN edge scatter: out[si[e]] += dinv[gi]*dinv[si] * t[gi[e]] ----------------
__global__ __launch_bounds__(256) void gcn_scatter_kernel(
    const float* __restrict__ t, const int* __restrict__ gi, const int* __restrict__ si,
    const float* __restrict__ dinv, float* __restrict__ out, int E) {
  int tid = blockIdx.x * 256 + threadIdx.x;
  int e = tid >> 5, lane = tid & 31;
  if (e >= E) return;
  int s = gi[e], d = si[e];
  float norm = dinv[s] * dinv[d];
  float4 v = *(const float4*)(t + (size_t)s * F + lane * 4);
  float* o = out + (size_t)d * F + lane * 4;
  atomicAdd(o + 0, norm * v.x);
  atomicAdd(o + 1, norm * v.y);
  atomicAdd(o + 2, norm * v.z);
  atomicAdd(o + 3, norm * v.w);
}

// ---------------- h = relu(0.5*(a_out + b_out) + 0.5*(a_in + b_in)) ----------------
__global__ void combine_kernel(const float* __restrict__ a_in, const float* __restrict__ a_out,
                               const float* __restrict__ b_in, const float* __restrict__ b_out,
                               float* __restrict__ h, int total) {
  int i = blockIdx.x * blockDim.x + threadIdx.x;
  if (i >= total) return;
  int c = i & (F - 1);
  float v = 0.5f * (a_out[i] + b_out[c]) + 0.5f * (a_in[i] + b_in[c]);
  h[i] = v > 0.f ? v : 0.f;
}

// ---------------- GATv2 pass A: per-edge logits + segment max ----------------
__global__ __launch_bounds__(256) void gat_logits_kernel(
    const float* __restrict__ gl, const float* __restrict__ gr,
    const int* __restrict__ src, const int* __restrict__ dst,
    const float* __restrict__ ea, const float* __restrict__ we,
    const float* __restrict__ att, float* __restrict__ logits,
    float* __restrict__ m, int E, int N) {
  int tid = blockIdx.x * 256 + threadIdx.x;
  int e = tid >> 5, lane = tid & 31;
  if (e >= E + N) return;
  int s, d; float w;
  if (e < E) { s = src[e]; d = dst[e]; w = ea[e]; }
  else       { s = e - E;  d = s;      w = 1.0f; }
  float4 a = *(const float4*)(gl + (size_t)s * F + lane * 4);
  float4 b = *(const float4*)(gr + (size_t)d * F + lane * 4);
  float4 c = *(const float4*)(we + lane * 4);
  float4 t = *(const float4*)(att + lane * 4);
  float v0 = a.x + b.x + w * c.x; v0 = v0 > 0.f ? v0 : 0.2f * v0;
  float v1 = a.y + b.y + w * c.y; v1 = v1 > 0.f ? v1 : 0.2f * v1;
  float v2 = a.z + b.z + w * c.z; v2 = v2 > 0.f ? v2 : 0.2f * v2;
  float v3 = a.w + b.w + w * c.w; v3 = v3 > 0.f ? v3 : 0.2f * v3;
  float p = v0 * t.x + v1 * t.y + v2 * t.z + v3 * t.w;
  // lanes 0-7 hold head 0 (channels 0-31), lanes 8-15 head 1, etc.
  p += __shfl_xor(p, 1, 32);
  p += __shfl_xor(p, 2, 32);
  p += __shfl_xor(p, 4, 32);
  if ((lane & 7) == 0) {
    int h = lane >> 3;
    logits[(size_t)e * 4 + h] = p;
    atomicMaxFloat(m + (size_t)d * 4 + h, p);
  }
}

// ---------------- GATv2 pass B: exp + segment sum ----------------
__global__ void gat_exp_kernel(float* __restrict__ logits, const float* __restrict__ m,
                               float* __restrict__ denom, const int* __restrict__ dst,
                               int E, int N) {
  int idx = blockIdx.x * blockDim.x + threadIdx.x;
  if (idx >= (E + N) * 4) return;
  int e = idx >> 2, h = idx & 3;
  int d = (e < E) ? dst[e] : (e - E);
  float ex = expf(logits[idx] - m[d * 4 + h]);
  logits[idx] = ex;
  atomicAdd(denom + d * 4 + h, ex);
}

// ---------------- GATv2 pass C: weighted scatter ----------------
__global__ __launch_bounds__(256) void gat_scatter_kernel(
    const float* __restrict__ gl, const float* __restrict__ ex,
    const float* __restrict__ denom, const int* __restrict__ src,
    const int* __restrict__ dst, float* __restrict__ out2, int E, int N) {
  int tid = blockIdx.x * 256 + threadIdx.x;
  int e = tid >> 5, lane = tid & 31;
  if (e >= E + N) return;
  int s, d;
  if (e < E) { s = src[e]; d = dst[e]; }
  else       { s = e - E;  d = s; }
  int h = lane >> 3;
  float aw = ex[(size_t)e * 4 + h] / denom[(size_t)d * 4 + h];
  float4 v = *(const float4*)(gl + (size_t)s * F + lane * 4);
  float* o = out2 + (size_t)d * F + lane * 4;
  atomicAdd(o + 0, aw * v.x);
  atomicAdd(o + 1, aw * v.y);
  atomicAdd(o + 2, aw * v.z);
  atomicAdd(o + 3, aw * v.w);
}

// ---------------- graph mean-pool accumulation ----------------
__global__ __launch_bounds__(256) void pool_kernel(
    const float* __restrict__ h2, const float* __restrict__ out2,
    const float* __restrict__ gat_b, const int* __restrict__ batch,
    float* __restrict__ zsum, float* __restrict__ cnt, int N) {
  int tid = blockIdx.x * 256 + threadIdx.x;
  int n = tid >> 7, j = tid & 127;
  if (n >= N) return;
  int g = batch[n];
  atomicAdd(zsum + (size_t)g * 256 + j, h2[(size_t)n * F + j]);
  atomicAdd(zsum + (size_t)g * 256 + 128 + j, out2[(size_t)n * F + j] + gat_b[j]);
  if (j == 0) atomicAdd(cnt + g, 1.0f);
}

// ---------------- MLP head ----------------
__global__ __launch_bounds__(128) void head_kernel(
    const float* __restrict__ zsum, const float* __restrict__ cnt,
    const float* __restrict__ w1, const float* __restrict__ b1,
    const float* __restrict__ w2, const float* __restrict__ b2,
    float* __restrict__ out) {
  int g = blockIdx.x, j = threadIdx.x;
  __shared__ float hid[128];
  float inv = 1.0f / fmaxf(cnt[g], 1.0f);
  float acc = b1[j];
  for (int k = 0; k < 256; ++k) acc += (zsum[(size_t)g * 256 + k] * inv) * w1[k * 128 + j];
  hid[j] = fmaxf(acc, 0.f);
  __syncthreads();
  if (j < 2) {
    float o = b2[j];
    for (int k = 0; k < 128; ++k) o += hid[k] * w2[k * 2 + j];
    out[g * 2 + j] = o;
  }
}

extern "C" void kernel_launch(void* const* d_in, const int* in_sizes, int n_in,
                              void* d_out, int out_size, void* d_ws, size_t ws_size,
                              hipStream_t stream) {
  (void)n_in; (void)out_size; (void)ws_size;
  const float* x        = (const float*)d_in[0];
  const int*   ei       = (const int*)d_in[1];
  const float* ea       = (const float*)d_in[2];
  const int*   batch    = (const int*)d_in[3];
  const float* dir_w_in = (const float*)d_in[4];
  const float* dir_b_in = (const float*)d_in[5];
  const float* dir_w_out= (const float*)d_in[6];
  const float* dir_b_out= (const float*)d_in[7];
  const float* gat_wl   = (const float*)d_in[8];
  const float* gat_wr   = (const float*)d_in[9];
  const float* gat_we   = (const float*)d_in[10];
  const float* gat_att  = (const float*)d_in[11];
  const float* gat_b    = (const float*)d_in[12];
  const float* w1       = (const float*)d_in[13];
  const float* b1       = (const float*)d_in[14];
  const float* w2       = (const float*)d_in[15];
  const float* b2       = (const float*)d_in[16];

  const int N = in_sizes[0] / F;
  const int E = in_sizes[1] / 2;
  const int* src = ei;
  const int* dst = ei + E;

  const size_t NF = (size_t)N * F;
  float* p = (float*)d_ws;
  float* gl     = p; p += NF;
  float* gr     = p; p += NF;
  float* t0     = p; p += NF;
  float* t1     = p; p += NF;
  float* a0     = p; p += NF;
  float* a1     = p; p += NF;
  float* hb     = p; p += NF;
  float* out2   = p; p += NF;
  float* logits = p; p += (size_t)(E + N) * 4;
  float* mbuf   = p; p += (size_t)N * 4;
  float* denom  = p; p += (size_t)N * 4;
  float* din    = p; p += N;
  float* dou    = p; p += N;
  float* zsum   = p; p += (size_t)G_NUM * 256;
  float* cnt    = p; p += G_NUM;

  // zero/init scratch (graph-capture safe)
  hipMemsetAsync(a0, 0, NF * sizeof(float), stream);
  hipMemsetAsync(a1, 0, NF * sizeof(float), stream);
  hipMemsetAsync(out2, 0, NF * sizeof(float), stream);
  hipMemsetAsync(denom, 0, (size_t)N * 4 * sizeof(float), stream);
  hipMemsetAsync(din, 0, (size_t)N * sizeof(float), stream);
  hipMemsetAsync(dou, 0, (size_t)N * sizeof(float), stream);
  hipMemsetAsync(zsum, 0, ((size_t)G_NUM * 256 + G_NUM) * sizeof(float), stream);
  fill_kernel<<<(N * 4 + 255) / 256, 256, 0, stream>>>(mbuf, -INFINITY, N * 4);

  // degrees + normalization
  degree_kernel<<<(E + 255) / 256, 256, 0, stream>>>(src, dst, din, dou, E);
  dinv_kernel<<<(N + 255) / 256, 256, 0, stream>>>(din, dou, N);

  const int gb   = (N + 63) / 64;
  const int eb   = (E * 32 + 255) / 256;          // wave-per-edge grids
  const int enb  = ((E + N) * 32 + 255) / 256;
  const int nfb  = (int)((NF + 255) / 256);

  // GEMMs on x (GAT transforms + DirGNN layer 0)
  gemm128_wmma<<<gb, 256, 0, stream>>>(x, gat_wl, gl, N);
  gemm128_wmma<<<gb, 256, 0, stream>>>(x, gat_wr, gr, N);
  gemm128_wmma<<<gb, 256, 0, stream>>>(x, dir_w_in, t0, N);
  gemm128_wmma<<<gb, 256, 0, stream>>>(x, dir_w_out, t1, N);

  // DirGNN layer 0
  gcn_scatter_kernel<<<eb, 256, 0, stream>>>(t0, src, dst, din, a0, E);
  gcn_scatter_kernel<<<eb, 256, 0, stream>>>(t1, dst, src, dou, a1, E);
  combine_kernel<<<nfb, 256, 0, stream>>>(a0, a1, dir_b_in, dir_b_out, hb, (int)NF);

  // DirGNN layer 1
  hipMemsetAsync(a0, 0, NF * sizeof(float), stream);
  hipMemsetAsync(a1, 0, NF * sizeof(float), stream);
  gemm128_wmma<<<gb, 256, 0, stream>>>(hb, dir_w_in + F * F, t0, N);
  gemm128_wmma<<<gb, 256, 0, stream>>>(hb, dir_w_out + F * F, t1, N);
  gcn_scatter_kernel<<<eb, 256, 0, stream>>>(t0, src, dst, din, a0, E);
  gcn_scatter_kernel<<<eb, 256, 0, stream>>>(t1, dst, src, dou, a1, E);
  combine_kernel<<<nfb, 256, 0, stream>>>(a0, a1, dir_b_in + F, dir_b_out + F, hb, (int)NF);

  // GATv2 branch (edges + self loops)
  gat_logits_kernel<<<enb, 256, 0, stream>>>(gl, gr, src, dst, ea, gat_we, gat_att,
                                             logits, mbuf, E, N);
  gat_exp_kernel<<<((E + N) * 4 + 255) / 256, 256, 0, stream>>>(logits, mbuf, denom, dst, E, N);
  gat_scatter_kernel<<<enb, 256, 0, stream>>>(gl, logits, denom, src, dst, out2, E, N);

  // pooling + head
  pool_kernel<<<nfb, 256, 0, stream>>>(hb, out2, gat_b, batch, zsum, cnt, N);
  head_kernel<<<G_NUM, 128, 0, stream>>>(zsum, cnt, w1, b1, w2, b2, (float*)d_out);
}